// GAT_50397146251359
// MI455X (gfx1250) — compile-verified
//
#include <hip/hip_runtime.h>
#include <hip/hip_bf16.h>

// ---------------------------------------------------------------------------
// GATv2 (2 layers) + mean-pool + MLP head for gfx1250 (MI455X, wave32).
// Dense transforms use V_WMMA_F32_16X16X4_F32 (exact fp32 matrix cores).
// Edge phase is wave-per-edge (lane == channel, DH == 32 == wave32).
// ---------------------------------------------------------------------------

typedef __attribute__((ext_vector_type(2))) float v2f;
typedef __attribute__((ext_vector_type(8))) float v8f;

#define DH   32
#define H1   8
#define H2   2
#define GNUM 64
#define DOUT 10

// ---------------------------------------------------------------------------
// fp32 WMMA GEMM: C[M,Nc] = A[M,K] * B[K,Nc].  One wave -> 16x64 C tile.
// A frag (16x4):  lanes 0-15 hold M=lane, K={0,1}; lanes 16-31 K={2,3}.
// B frag (4x16):  lanes 0-15 hold N=lane, K={0,1}; lanes 16-31 K={2,3}.
// C (16x16):      VGPR r: lanes0-15 -> M=r, lanes16-31 -> M=r+8, N=lane&15.
// ---------------------------------------------------------------------------
__global__ __launch_bounds__(128)
void wmma_gemm_f32(const float* __restrict__ A, const float* __restrict__ B,
                   float* __restrict__ C, int M, int K, int Nc)
{
    const int lane  = threadIdx.x;                       // 0..31
    const int tileM = blockIdx.x * blockDim.y + threadIdx.y;
    const int tileN = blockIdx.y * 64;
    if (tileM * 16 >= M) return;                         // uniform per wave
    const int m  = lane & 15;
    const int kh = lane >> 4;                            // K half select
    const long long row = (long long)tileM * 16 + m;

    v8f acc0 = {}, acc1 = {}, acc2 = {}, acc3 = {};

    #pragma unroll 4
    for (int k0 = 0; k0 < K; k0 += 4) {
        v2f a;
        const float* Ar = A + row * K + k0 + 2 * kh;
        a.x = Ar[0];
        a.y = Ar[1];

        const float* B0 = B + (long long)(k0 + 2 * kh) * Nc + tileN + m;
        const float* B1 = B0 + Nc;
        v2f b0, b1, b2, b3;
        b0.x = B0[0];  b0.y = B1[0];
        b1.x = B0[16]; b1.y = B1[16];
        b2.x = B0[32]; b2.y = B1[32];
        b3.x = B0[48]; b3.y = B1[48];

        acc0 = __builtin_amdgcn_wmma_f32_16x16x4_f32(false, a, false, b0, (short)0, acc0, false, false);
        acc1 = __builtin_amdgcn_wmma_f32_16x16x4_f32(false, a, false, b1, (short)0, acc1, false, false);
        acc2 = __builtin_amdgcn_wmma_f32_16x16x4_f32(false, a, false, b2, (short)0, acc2, false, false);
        acc3 = __builtin_amdgcn_wmma_f32_16x16x4_f32(false, a, false, b3, (short)0, acc3, false, false);
    }

    #pragma unroll
    for (int r = 0; r < 8; ++r) {
        const long long orow = (long long)tileM * 16 + r + 8 * kh;
        float* Cr = C + orow * Nc + tileN + m;
        Cr[0]  = acc0[r];
        Cr[16] = acc1[r];
        Cr[32] = acc2[r];
        Cr[48] = acc3[r];
    }
}

// ---------------------------------------------------------------------------
// float atomic max via monotone int/uint bit trick (init must be -inf)
// ---------------------------------------------------------------------------
__device__ __forceinline__ void atomicMaxFloat(float* addr, float value)
{
    if (value >= 0.0f)
        atomicMax((int*)addr, __float_as_int(value));
    else
        atomicMin((unsigned int*)addr, (unsigned int)__float_as_int(value));
}

// ---------------------------------------------------------------------------
// Edge scores: e[e,h] = att[h,:] . leaky_relu(xl[src,h,:] + xr[dst,h,:])
// plus segment-max into mbuf.  One wave per edge, lane = channel.
// ---------------------------------------------------------------------------
__global__ __launch_bounds__(256)
void gat_edge_scores(const float* __restrict__ xl, const float* __restrict__ xr,
                     const int* __restrict__ srcA, const int* __restrict__ dstA,
                     int E0, int Etot, const float* __restrict__ att, int H,
                     float* __restrict__ escore, float* __restrict__ mbuf)
{
    const int gtid = blockIdx.x * blockDim.x + threadIdx.x;
    const int e    = gtid >> 5;
    const int lane = gtid & 31;
    if (e >= Etot) return;
    const int s = (e < E0) ? srcA[e] : (e - E0);
    const int d = (e < E0) ? dstA[e] : (e - E0);
    const int stride = H * DH;
    for (int h = 0; h < H; ++h) {
        float v = xl[(long long)s * stride + h * DH + lane] +
                  xr[(long long)d * stride + h * DH + lane];
        v = v > 0.0f ? v : 0.2f * v;                     // leaky_relu 0.2
        v *= att[h * DH + lane];
        #pragma unroll
        for (int off = 16; off > 0; off >>= 1)
            v += __shfl_xor(v, off, 32);
        if (lane == 0) {
            escore[(long long)e * H + h] = v;
            atomicMaxFloat(&mbuf[(long long)d * H + h], v);
        }
    }
}

// p = exp(e - m[dst]); z[dst] += p  (escore updated in place)
__global__ void gat_softmax_p(float* __restrict__ escore,
                              const int* __restrict__ dstA,
                              int E0, int Etot, int H,
                              const float* __restrict__ mbuf,
                              float* __restrict__ zbuf)
{
    const long long t = (long long)blockIdx.x * blockDim.x + threadIdx.x;
    if (t >= (long long)Etot * H) return;
    const int e = (int)(t / H);
    const int h = (int)(t % H);
    const int d = (e < E0) ? dstA[e] : (e - E0);
    const float p = __expf(escore[t] - mbuf[(long long)d * H + h]);
    escore[t] = p;
    atomicAdd(&zbuf[(long long)d * H + h], p);
}

// agg[dst] += (p/z[dst]) * xl[src]   (wave per edge, lane = channel)
__global__ __launch_bounds__(256)
void gat_aggregate(const float* __restrict__ xl, const float* __restrict__ p,
                   const int* __restrict__ srcA, const int* __restrict__ dstA,
                   int E0, int Etot, int H,
                   const float* __restrict__ zbuf, float* __restrict__ agg)
{
    const int gtid = blockIdx.x * blockDim.x + threadIdx.x;
    const int e    = gtid >> 5;
    const int lane = gtid & 31;
    if (e >= Etot) return;
    const int s = (e < E0) ? srcA[e] : (e - E0);
    const int d = (e < E0) ? dstA[e] : (e - E0);
    const int stride = H * DH;
    for (int h = 0; h < H; ++h) {
        const float alpha = p[(long long)e * H + h] / zbuf[(long long)d * H + h];
        atomicAdd(&agg[(long long)d * stride + h * DH + lane],
                  alpha * xl[(long long)s * stride + h * DH + lane]);
    }
}

// h = elu(h + bias)   (in place)
__global__ void bias_elu(float* __restrict__ h, const float* __restrict__ bias,
                         long long total, int C)
{
    const long long t = (long long)blockIdx.x * blockDim.x + threadIdx.x;
    if (t >= total) return;
    const int c = (int)(t % C);
    const float v = h[t] + bias[c];
    h[t] = v > 0.0f ? v : (__expf(v) - 1.0f);
}

// sums[batch[i]] += agg[i] + bias; cnt[batch[i]] += 1
__global__ void pool_sum(const float* __restrict__ agg, const float* __restrict__ bias,
                         const int* __restrict__ batch, long long total, int C,
                         float* __restrict__ sums, float* __restrict__ cnt)
{
    const long long t = (long long)blockIdx.x * blockDim.x + threadIdx.x;
    if (t >= total) return;
    const int i = (int)(t / C);
    const int c = (int)(t % C);
    const int g = batch[i];
    atomicAdd(&sums[(long long)g * C + c], agg[t] + bias[c]);
    if (c == 0) atomicAdd(&cnt[g], 1.0f);
}

__global__ void pool_mean(float* __restrict__ sums, const float* __restrict__ cnt,
                          int total, int C)
{
    const int t = blockIdx.x * blockDim.x + threadIdx.x;
    if (t >= total) return;
    sums[t] = sums[t] / fmaxf(cnt[t / C], 1.0f);
}

// MLP head: fc1(64->32) relu, fc2(32->16) relu, fc(16->10). One thread/graph.
// out layout: c [G,10] followed by x2 [G,16].
__global__ void mlp_head(const float* __restrict__ hg,
                         const float* __restrict__ fc1w, const float* __restrict__ fc1b,
                         const float* __restrict__ fc2w, const float* __restrict__ fc2b,
                         const float* __restrict__ fcw,  const float* __restrict__ fcb,
                         float* __restrict__ out, int G)
{
    const int g = blockIdx.x * blockDim.x + threadIdx.x;
    if (g >= G) return;
    float x1[32];
    for (int j = 0; j < 32; ++j) {
        float acc = fc1b[j];
        for (int k = 0; k < 64; ++k) acc += hg[g * 64 + k] * fc1w[k * 32 + j];
        x1[j] = fmaxf(acc, 0.0f);
    }
    float x2[16];
    for (int j = 0; j < 16; ++j) {
        float acc = fc2b[j];
        for (int k = 0; k < 32; ++k) acc += x1[k] * fc2w[k * 16 + j];
        x2[j] = fmaxf(acc, 0.0f);
    }
    for (int j = 0; j < DOUT; ++j) {
        float acc = fcb[j];
        for (int k = 0; k < 16; ++k) acc += x2[k] * fcw[k * DOUT + j];
        out[g * DOUT + j] = acc;
    }
    for (int j = 0; j < 16; ++j) out[G * DOUT + g * 16 + j] = x2[j];
}

__global__ void fill_f32(float* __restrict__ p, float v, long long n)
{
    const long long t = (long long)blockIdx.x * blockDim.x + threadIdx.x;
    if (t < n) p[t] = v;
}

// ---------------------------------------------------------------------------
extern "C" void kernel_launch(void* const* d_in, const int* in_sizes, int n_in,
                              void* d_out, int out_size, void* d_ws, size_t ws_size,
                              hipStream_t stream)
{
    const float* x    = (const float*)d_in[0];
    const int*   ei   = (const int*)  d_in[1];
    const int*   bat  = (const int*)  d_in[2];
    const float* W1l  = (const float*)d_in[3];
    const float* W1r  = (const float*)d_in[4];
    const float* a1   = (const float*)d_in[5];
    const float* b1   = (const float*)d_in[6];
    const float* W2l  = (const float*)d_in[7];
    const float* W2r  = (const float*)d_in[8];
    const float* a2   = (const float*)d_in[9];
    const float* b2   = (const float*)d_in[10];
    const float* fc1w = (const float*)d_in[11];
    const float* fc1b = (const float*)d_in[12];
    const float* fc2w = (const float*)d_in[13];
    const float* fc2b = (const float*)d_in[14];
    const float* fcw  = (const float*)d_in[15];
    const float* fcb  = (const float*)d_in[16];
    float* out = (float*)d_out;

    const int N    = in_sizes[2];          // 30000 (multiple of 16)
    const int DIN  = in_sizes[0] / N;      // 128
    const int E0   = in_sizes[1] / 2;      // 480000
    const int Etot = E0 + N;               // + self loops
    const int C1   = H1 * DH;              // 256
    const int C2   = H2 * DH;              // 64
    const int* srcA = ei;
    const int* dstA = ei + E0;

    // ---- workspace layout (floats); layer 2 reuses layer-1 regions --------
    float* ws = (float*)d_ws;
    long long o = 0;
    float* xl1  = ws + o; o += (long long)N * C1;   // reused as xl2
    float* xr1  = ws + o; o += (long long)N * C1;   // reused as xr2
    float* e1   = ws + o; o += (long long)Etot * H1;// reused as e2
    float* m1   = ws + o; o += (long long)N * H1;   // reused as m2
    float* z1   = ws + o; o += (long long)N * H1;   // reused as z2
    float* agg1 = ws + o; o += (long long)N * C1;   // -> h1 -> agg2
    float* sums = ws + o; o += (long long)GNUM * C2;
    float* cnt  = ws + o; o += GNUM;

    const float NEG_INF = -__builtin_inff();
    #define FILL(ptr, val, n) fill_f32<<<dim3((unsigned)(((n) + 255) / 256)), 256, 0, stream>>>((ptr), (val), (long long)(n))

    // ================= layer 1 =================
    FILL(agg1, 0.0f, (long long)N * C1);
    FILL(z1,   0.0f, (long long)N * H1);
    FILL(m1, NEG_INF, (long long)N * H1);

    {   // xl1 = x @ W1l ; xr1 = x @ W1r
        dim3 blk(32, 4);
        dim3 grd((N / 16 + 3) / 4, C1 / 64);
        wmma_gemm_f32<<<grd, blk, 0, stream>>>(x, W1l, xl1, N, DIN, C1);
        wmma_gemm_f32<<<grd, blk, 0, stream>>>(x, W1r, xr1, N, DIN, C1);
    }
    {
        const long long thr = (long long)Etot * 32;
        gat_edge_scores<<<dim3((unsigned)((thr + 255) / 256)), 256, 0, stream>>>(
            xl1, xr1, srcA, dstA, E0, Etot, a1, H1, e1, m1);
        const long long pt = (long long)Etot * H1;
        gat_softmax_p<<<dim3((unsigned)((pt + 255) / 256)), 256, 0, stream>>>(
            e1, dstA, E0, Etot, H1, m1, z1);
        gat_aggregate<<<dim3((unsigned)((thr + 255) / 256)), 256, 0, stream>>>(
            xl1, e1, srcA, dstA, E0, Etot, H1, z1, agg1);
        const long long nt = (long long)N * C1;
        bias_elu<<<dim3((unsigned)((nt + 255) / 256)), 256, 0, stream>>>(agg1, b1, nt, C1);
    }

    // ================= layer 2 =================
    float* h1   = agg1;
    float* xl2  = xl1;
    float* xr2  = xr1;
    float* e2   = e1;
    float* m2   = m1;
    float* z2   = z1;
    {   // xl2 = h1 @ W2l ; xr2 = h1 @ W2r   (then h1 region becomes agg2)
        dim3 blk(32, 4);
        dim3 grd((N / 16 + 3) / 4, C2 / 64);
        wmma_gemm_f32<<<grd, blk, 0, stream>>>(h1, W2l, xl2, N, C1, C2);
        wmma_gemm_f32<<<grd, blk, 0, stream>>>(h1, W2r, xr2, N, C1, C2);
    }
    float* agg2 = agg1;  // h1 consumed; safe to reuse
    FILL(agg2, 0.0f, (long long)N * C2);
    FILL(z2,   0.0f, (long long)N * H2);
    FILL(m2, NEG_INF, (long long)N * H2);
    {
        const long long thr = (long long)Etot * 32;
        gat_edge_scores<<<dim3((unsigned)((thr + 255) / 256)), 256, 0, stream>>>(
            xl2, xr2, srcA, dstA, E0, Etot, a2, H2, e2, m2);
        const long long pt = (long long)Etot * H2;
        gat_softmax_p<<<dim3((unsigned)((pt + 255) / 256)), 256, 0, stream>>>(
            e2, dstA, E0, Etot, H2, m2, z2);
        gat_aggregate<<<dim3((unsigned)((thr + 255) / 256)), 256, 0, stream>>>(
            xl2, e2, srcA, dstA, E0, Etot, H2, z2, agg2);
    }

    // ================= pooling + MLP head =================
    FILL(sums, 0.0f, (long long)GNUM * C2);
    FILL(cnt,  0.0f, GNUM);
    {
        const long long nt = (long long)N * C2;
        pool_sum<<<dim3((unsigned)((nt + 255) / 256)), 256, 0, stream>>>(
            agg2, b2, bat, nt, C2, sums, cnt);
        pool_mean<<<dim3((GNUM * C2 + 255) / 256), 256, 0, stream>>>(
            sums, cnt, GNUM * C2, C2);
        mlp_head<<<dim3(1), 64, 0, stream>>>(
            sums, fc1w, fc1b, fc2w, fc2b, fcw, fcb, out, GNUM);
    }
    #undef FILL
}